// AttentionMapLayer_21629455303185
// MI455X (gfx1250) — compile-verified
//
#include <hip/hip_runtime.h>

// Problem shape (fixed by the reference)
#define B_  16
#define H_  300
#define W_  25
#define C_  512
#define C4_ 128   // C_/4 floats per float4-thread; power of two -> idx>>7

typedef __attribute__((ext_vector_type(2))) float v2f;
typedef __attribute__((ext_vector_type(4))) float v4f;
typedef __attribute__((ext_vector_type(8))) float v8f;

__device__ __forceinline__ float inv_l2(float m) {
    // rsqrt(max(m, 1e-12)) with one Newton step to match lax.rsqrt precision
    m = fmaxf(m, 1e-12f);
    float r = rsqrtf(m);
    r = r * (1.5f - 0.5f * m * r * r);
    return r;
}

// ---------------------------------------------------------------------------
// Kernel 0: per-batch inverse L2 norms of s_o (B,W) and t_o (B,H) -> ws[0..31]
// Tiny (16 rows of <=300 elems); one wave total.
// ---------------------------------------------------------------------------
__global__ void norms_kernel(const float* __restrict__ s_o,
                             const float* __restrict__ t_o,
                             float* __restrict__ norms) {
    int b = threadIdx.x;
    if (b >= B_) return;
    float ss = 0.f;
    for (int w = 0; w < W_; ++w) { float v = s_o[b * W_ + w]; ss += v * v; }
    float ts = 0.f;
    for (int h = 0; h < H_; ++h) { float v = t_o[b * H_ + h]; ts += v * v; }
    norms[b]       = inv_l2(ss);   // inv ||s_o[b]||
    norms[B_ + b]  = inv_l2(ts);   // inv ||t_o[b]||
}

// ---------------------------------------------------------------------------
// Kernel 1: attention map a[b,h,w] = t_n[b,h]*s_n[b,w] + roi[h,w]
// as a K=1 (padded to 4) GEMM per 16x16 tile using V_WMMA_F32_16X16X4_F32.
//   A (16x4 f32, 2 VGPRs): K=0 column = t_n rows, lanes 0-15 of VGPR0.
//   B (4x16 f32, 2 VGPRs): K=0 row    = s_n cols, lanes 0-15 of VGPR0.
//   C/D (16x16 f32, 8 VGPRs): row = v + 8*(lane>=16), col = lane&15.
// One wave (32 threads) per tile; grid = (ceil(W/16), ceil(H/16), B) = (2,19,16).
// ---------------------------------------------------------------------------
__global__ void amap_wmma_kernel(const float* __restrict__ s_o,
                                 const float* __restrict__ t_o,
                                 const float* __restrict__ roi,
                                 const float* __restrict__ norms,
                                 float* __restrict__ amap) {
    const int lane = threadIdx.x;          // 0..31 (wave32)
    const int wt   = blockIdx.x;           // W tile
    const int ht   = blockIdx.y;           // H tile
    const int b    = blockIdx.z;           // batch
    const int half = lane >> 4;            // 0: lanes 0-15, 1: lanes 16-31
    const int l    = lane & 15;

    const float inv_s = norms[b];
    const float inv_t = norms[B_ + b];

    const int h0 = ht * 16;
    const int w0 = wt * 16;

    // A matrix: t_n in K=0 slot (VGPR0, lanes 0-15); everything else zero.
    v2f A; A.x = 0.f; A.y = 0.f;
    if (half == 0) {
        int h = h0 + l;
        A.x = (h < H_) ? t_o[b * H_ + h] * inv_t : 0.f;
    }
    // B matrix: s_n in row K=0 (VGPR0, lanes 0-15); everything else zero.
    v2f Bm; Bm.x = 0.f; Bm.y = 0.f;
    if (half == 0) {
        int w = w0 + l;
        Bm.x = (w < W_) ? s_o[b * W_ + w] * inv_s : 0.f;
    }
    // C accumulator: roi tile (zero-padded out of range).
    v8f c;
#pragma unroll
    for (int v = 0; v < 8; ++v) {
        int h = h0 + half * 8 + v;
        int w = w0 + l;
        c[v] = (h < H_ && w < W_) ? roi[h * W_ + w] : 0.f;
    }

    // D = A x B + C  (full EXEC here; divergence above has reconverged)
    v8f d = __builtin_amdgcn_wmma_f32_16x16x4_f32(
        /*neg_a=*/false, A, /*neg_b=*/false, Bm,
        /*c_mod=*/(short)0, c, /*reuse_a=*/false, /*reuse_b=*/false);

#pragma unroll
    for (int v = 0; v < 8; ++v) {
        int h = h0 + half * 8 + v;
        int w = w0 + l;
        if (h < H_ && w < W_)
            amap[(b * H_ + h) * W_ + w] = d[v];
    }
}

// ---------------------------------------------------------------------------
// Kernel 2: bandwidth-bound stream: out = amap[bhw] * ipt, float4 per thread.
// NT hints on the 2x245MB stream (won't fit 192MB L2); amap reads stay cached.
// ---------------------------------------------------------------------------
__global__ void scale_stream_kernel(const float* __restrict__ ipt,
                                    const float* __restrict__ amap,
                                    float* __restrict__ out, int n4) {
    int idx = blockIdx.x * blockDim.x + threadIdx.x;
    if (idx >= n4) return;
    float s = amap[idx >> 7];                       // C4_ == 128
    const v4f* in4 = (const v4f*)ipt;
    v4f v = __builtin_nontemporal_load(in4 + idx);
    v4f r = v * s;
    __builtin_nontemporal_store(r, (v4f*)out + idx);
}

// ---------------------------------------------------------------------------
// Fallback if ws can't hold the 480KB map: compute scale inline per thread.
// Extra loads are tiny (s_o/t_o/roi/norms all L2-resident), still BW-bound.
// ---------------------------------------------------------------------------
__global__ void fused_stream_kernel(const float* __restrict__ ipt,
                                    const float* __restrict__ s_o,
                                    const float* __restrict__ t_o,
                                    const float* __restrict__ roi,
                                    const float* __restrict__ norms,
                                    float* __restrict__ out, int n4) {
    int idx = blockIdx.x * blockDim.x + threadIdx.x;
    if (idx >= n4) return;
    int bhw = idx >> 7;
    int w  = bhw % W_;
    int bh = bhw / W_;
    int h  = bh % H_;
    int b  = bh / H_;
    float s = s_o[b * W_ + w] * norms[b] * t_o[b * H_ + h] * norms[B_ + b]
            + roi[h * W_ + w];
    const v4f* in4 = (const v4f*)ipt;
    v4f v = __builtin_nontemporal_load(in4 + idx);
    v4f r = v * s;
    __builtin_nontemporal_store(r, (v4f*)out + idx);
}

extern "C" void kernel_launch(void* const* d_in, const int* in_sizes, int n_in,
                              void* d_out, int out_size, void* d_ws, size_t ws_size,
                              hipStream_t stream) {
    const float* s_o = (const float*)d_in[0];   // (16, 25)
    const float* t_o = (const float*)d_in[1];   // (16, 300)
    const float* ipt = (const float*)d_in[2];   // (16, 300, 25, 512)
    const float* roi = (const float*)d_in[3];   // (1, 300, 25)
    float* out = (float*)d_out;
    float* wsf = (float*)d_ws;

    const int n4 = out_size / 4;                // float4 elements
    const size_t need = (size_t)(2 * B_) * sizeof(float)
                      + (size_t)B_ * H_ * W_ * sizeof(float);

    // Per-batch inverse norms -> ws[0..31]
    norms_kernel<<<1, 32, 0, stream>>>(s_o, t_o, wsf);

    if (ws_size >= need) {
        float* amap = wsf + 2 * B_;
        dim3 gmap((W_ + 15) / 16, (H_ + 15) / 16, B_);   // (2, 19, 16)
        amap_wmma_kernel<<<gmap, 32, 0, stream>>>(s_o, t_o, roi, wsf, amap);
        scale_stream_kernel<<<(n4 + 255) / 256, 256, 0, stream>>>(ipt, amap, out, n4);
    } else {
        fused_stream_kernel<<<(n4 + 255) / 256, 256, 0, stream>>>(
            ipt, s_o, t_o, roi, wsf, out, n4);
    }
}